// DepthlessTransformer_79912161509773
// MI455X (gfx1250) — compile-verified
//
#include <hip/hip_runtime.h>
#include <math.h>

// ---------------------------------------------------------------------------
// DepthlessTransformer forward for gfx1250 (MI455X)
//   - all GEMMs + attention matmuls: V_WMMA_F32_16X16X4_F32 (fp32-exact path)
//   - all LDS tile staging: Tensor Data Mover (TENSOR_LOAD_TO_LDS) with
//     hardware LDS padding for bank-conflict-free fragment reads
// ---------------------------------------------------------------------------

typedef __attribute__((ext_vector_type(2))) float v2f;
typedef __attribute__((ext_vector_type(8))) float v8f;
typedef __attribute__((ext_vector_type(4))) unsigned int u32x4;
typedef __attribute__((ext_vector_type(4))) int i32x4;
typedef __attribute__((ext_vector_type(8))) int i32x8;

#define DIMD   512
#define NSEQ   256
#define NBATCH 2
#define NBLK   6
#define NHEAD  8
#define DHEAD  64
#define DFF    1365
#define RTOT   3072          // 6*2*256 rows
#define EPSF   1.1920929e-07f

__device__ __forceinline__ v8f wmma_f32x4(v2f a, v2f b, v8f c) {
  return __builtin_amdgcn_wmma_f32_16x16x4_f32(false, a, false, b, (short)0, c, false, false);
}

// ---------------------------------------------------------------------------
// TDM: 2D tile load Global -> LDS (fp32 elements).
//   lds_bytes      : byte offset of destination inside workgroup LDS
//   gtile          : global address of the TILE start
//   tw, th         : remaining tensor extent (elements) for OOB zero-fill
//   tile_w, tile_h : tile dims in elements (tile_w*4B must be mult of 4B)
//   stride_elems   : tensor dim0 (row) stride in elements
//   pad_code       : pad interval code (interval = 2<<code DWORDs)
//   pad_m1         : pad amount in DWORDs, minus 1
// D# layout per CDNA5 ISA ch.8 (group0 128b, group1 256b; groups 2/3 zero = 2D).
// ---------------------------------------------------------------------------
__device__ __forceinline__ void tdm_load_2d(unsigned lds_bytes, const void* gtile,
                                            unsigned tw, unsigned th,
                                            unsigned tile_w, unsigned tile_h,
                                            unsigned stride_elems,
                                            unsigned pad_code, unsigned pad_m1) {
  const unsigned long long ga = (unsigned long long)(size_t)gtile;
  u32x4 g0;
  g0[0] = 1u;                                            // count=1, user mode
  g0[1] = lds_bytes;                                     // lds_addr
  g0[2] = (unsigned)(ga & 0xFFFFFFFFu);                  // global_addr[31:0]
  g0[3] = (unsigned)((ga >> 32) & 0x1FFFFFFu) | (2u << 30); // ga[56:32] | type=2
  i32x8 g1;
  g1[0] = (int)((2u << 16) | (1u << 20) | (pad_code << 22) | (pad_m1 << 25));
  g1[1] = (int)((tw & 0xFFFFu) << 16);                   // tensor_dim0[15:0]
  g1[2] = (int)((tw >> 16) | ((th & 0xFFFFu) << 16));    // dim0[31:16] | dim1[15:0]
  g1[3] = (int)((th >> 16) | (tile_w << 16));            // dim1[31:16] | tile_dim0
  g1[4] = (int)(tile_h & 0xFFFFu);                       // tile_dim1 (tile_dim2=0)
  g1[5] = (int)stride_elems;                             // tensor_dim0_stride[31:0]
  g1[6] = 0;
  g1[7] = 0;
  const i32x4 z4 = {0, 0, 0, 0};
#if defined(__clang_major__) && (__clang_major__ >= 23)
  const i32x8 z8 = {0, 0, 0, 0, 0, 0, 0, 0};
  __builtin_amdgcn_tensor_load_to_lds(g0, g1, z4, z4, z8, 0);
#else
  __builtin_amdgcn_tensor_load_to_lds(g0, g1, z4, z4, 0);
#endif
}

__device__ __forceinline__ float half16_max(float v) {
  v = fmaxf(v, __shfl_xor(v, 1, 32));
  v = fmaxf(v, __shfl_xor(v, 2, 32));
  v = fmaxf(v, __shfl_xor(v, 4, 32));
  v = fmaxf(v, __shfl_xor(v, 8, 32));
  return v;
}
__device__ __forceinline__ float half16_sum(float v) {
  v += __shfl_xor(v, 1, 32);
  v += __shfl_xor(v, 2, 32);
  v += __shfl_xor(v, 4, 32);
  v += __shfl_xor(v, 8, 32);
  return v;
}
__device__ __forceinline__ float wave32_sum(float v) {
  v += __shfl_xor(v, 1, 32);
  v += __shfl_xor(v, 2, 32);
  v += __shfl_xor(v, 4, 32);
  v += __shfl_xor(v, 8, 32);
  v += __shfl_xor(v, 16, 32);
  return v;
}

// ---------------------------------------------------------------------------
// C[R,N] = A[R,K] * W[N,K]^T (+ bias[N]).
// Block: 256 thr = 8 waves; block tile 64x128; wave tile 32x32 (2x2 WMMA tiles)
// -> each LDS fragment is reused twice (1 ds op per wmma).
// Tiles staged by TDM with pad: stride 33 floats (interval 32 dw, pad 1 dw).
// ---------------------------------------------------------------------------
__global__ __launch_bounds__(256)
void gemm_wmma(const float* __restrict__ A, const float* __restrict__ W,
               const float* __restrict__ bias, float* __restrict__ C,
               int R, int N, int K) {
  __shared__ float As[64][33];
  __shared__ float Ws[128][33];
  const int tid  = threadIdx.x;
  const int wid  = tid >> 5;
  const int lane = tid & 31;
  const int l16  = lane & 15;
  const int half = lane >> 4;
  const int rowBase = blockIdx.y * 64;
  const int colBase = blockIdx.x * 128;
  const int wr = (wid & 1) * 32;        // wave row offset in block tile
  const int wc = (wid >> 1) * 32;       // wave col offset in block tile
  const unsigned as_off = (unsigned)(size_t)&As[0][0];
  const unsigned ws_off = (unsigned)(size_t)&Ws[0][0];

  const v8f vzero = {0.f, 0.f, 0.f, 0.f, 0.f, 0.f, 0.f, 0.f};
  v8f acc[2][2];
#pragma unroll
  for (int i = 0; i < 2; ++i)
#pragma unroll
    for (int j = 0; j < 2; ++j) acc[i][j] = vzero;

  for (int k0 = 0; k0 < K; k0 += 32) {
    __syncthreads();
    if (wid == 0) {
      tdm_load_2d(as_off, A + (size_t)rowBase * K + k0,
                  (unsigned)(K - k0), (unsigned)(R - rowBase), 32, 64,
                  (unsigned)K, /*interval 32dw*/4, /*pad 1dw*/0);
      __builtin_amdgcn_s_wait_tensorcnt(0);
    } else if (wid == 1) {
      tdm_load_2d(ws_off, W + (size_t)colBase * K + k0,
                  (unsigned)(K - k0), (unsigned)(N - colBase), 32, 128,
                  (unsigned)K, 4, 0);
      __builtin_amdgcn_s_wait_tensorcnt(0);
    }
    __syncthreads();
#pragma unroll
    for (int kk = 0; kk < 32; kk += 4) {
      const int ka = kk + 2 * half;      // fragment layout: K = 2*lanehalf + vgpr
      v2f a0, a1, b0, b1;
      a0.x = As[wr + l16][ka];       a0.y = As[wr + l16][ka + 1];
      a1.x = As[wr + 16 + l16][ka];  a1.y = As[wr + 16 + l16][ka + 1];
      b0.x = Ws[wc + l16][ka];       b0.y = Ws[wc + l16][ka + 1];
      b1.x = Ws[wc + 16 + l16][ka];  b1.y = Ws[wc + 16 + l16][ka + 1];
      acc[0][0] = wmma_f32x4(a0, b0, acc[0][0]);
      acc[0][1] = wmma_f32x4(a0, b1, acc[0][1]);
      acc[1][0] = wmma_f32x4(a1, b0, acc[1][0]);
      acc[1][1] = wmma_f32x4(a1, b1, acc[1][1]);
    }
  }

#pragma unroll
  for (int ti = 0; ti < 2; ++ti)
#pragma unroll
    for (int tj = 0; tj < 2; ++tj)
#pragma unroll
      for (int j = 0; j < 8; ++j) {     // C layout: vgpr j -> row j (+8 hi half)
        const int r = rowBase + wr + ti * 16 + j + 8 * half;
        const int n = colBase + wc + tj * 16 + l16;
        if (r < R && n < N) {
          float v = acc[ti][tj][j];
          if (bias) v += bias[n];
          C[(size_t)r * N + n] = v;
        }
      }
}

// ---------------------------------------------------------------------------
// Self-attention (no scale, per reference): per (bb=0..11, head, 64-row block).
// Flash-style over 4 key chunks of 64, online softmax, WMMA for S and P*V.
// K/V chunks staged by TDM with pad: stride 68 floats (interval 64 dw, pad 4 dw).
// ---------------------------------------------------------------------------
__global__ __launch_bounds__(128)
void attn_self(const float* __restrict__ Q, const float* __restrict__ KV,
               float* __restrict__ O) {
  __shared__ float kl[64][68];
  __shared__ float vl[64][68];
  __shared__ float pl[4][16][68];

  const int idx  = blockIdx.x;          // 12*8*4 blocks
  const int rb   = idx & 3;
  const int head = (idx >> 2) & 7;
  const int bb   = idx >> 5;
  const int tid  = threadIdx.x;
  const int wid  = tid >> 5;
  const int lane = tid & 31;
  const int l16  = lane & 15;
  const int half = lane >> 4;
  const int rowTile = rb * 64 + wid * 16;
  const unsigned kl_off = (unsigned)(size_t)&kl[0][0];
  const unsigned vl_off = (unsigned)(size_t)&vl[0][0];

  const float* qrow = Q + ((size_t)(bb * NSEQ + rowTile + l16)) * DIMD + head * DHEAD;
  v2f qa[16];
#pragma unroll
  for (int t = 0; t < 16; ++t) {
    const int k = 4 * t + 2 * half;
    qa[t].x = qrow[k];
    qa[t].y = qrow[k + 1];
  }

  const v8f vzero = {0.f, 0.f, 0.f, 0.f, 0.f, 0.f, 0.f, 0.f};
  v8f of[4];
#pragma unroll
  for (int ft = 0; ft < 4; ++ft) of[ft] = vzero;
  float mi[8], li[8];
#pragma unroll
  for (int j = 0; j < 8; ++j) { mi[j] = -3.402823466e+38f; li[j] = 0.f; }

  for (int c = 0; c < 4; ++c) {
    __syncthreads();
    if (wid == 0) {
      tdm_load_2d(kl_off, KV + ((size_t)(bb * NSEQ + c * 64)) * 1024 + head * DHEAD,
                  64, 64, 64, 64, 1024, /*interval 64dw*/5, /*pad 4dw*/3);
      __builtin_amdgcn_s_wait_tensorcnt(0);
    } else if (wid == 1) {
      tdm_load_2d(vl_off, KV + ((size_t)(bb * NSEQ + c * 64)) * 1024 + 512 + head * DHEAD,
                  64, 64, 64, 64, 1024, 5, 3);
      __builtin_amdgcn_s_wait_tensorcnt(0);
    }
    __syncthreads();

    // S = q * k^T for this chunk: 4 column tiles, K = 64 features
    v8f sc[4];
#pragma unroll
    for (int ct = 0; ct < 4; ++ct) {
      v8f s = vzero;
#pragma unroll
      for (int t = 0; t < 16; ++t) {
        const int ka = 4 * t + 2 * half;
        v2f b;
        b.x = kl[ct * 16 + l16][ka];      // B[kfeat, nkey] = k[nkey, kfeat]
        b.y = kl[ct * 16 + l16][ka + 1];
        s = wmma_f32x4(qa[t], b, s);
      }
      sc[ct] = s;
    }

    // Online softmax (row = j + 8*half, cols striped over 16 lanes)
#pragma unroll
    for (int j = 0; j < 8; ++j) {
      float mrow = fmaxf(fmaxf(sc[0][j], sc[1][j]), fmaxf(sc[2][j], sc[3][j]));
      mrow = half16_max(mrow);
      const float mn = fmaxf(mi[j], mrow);
      const float scale = __expf(mi[j] - mn);
      li[j] *= scale;
#pragma unroll
      for (int ft = 0; ft < 4; ++ft) of[ft][j] *= scale;
      float psum = 0.f;
#pragma unroll
      for (int ct = 0; ct < 4; ++ct) {
        const float e = __expf(sc[ct][j] - mn);
        sc[ct][j] = e;
        psum += e;
      }
      li[j] += half16_sum(psum);
      mi[j] = mn;
    }

    // Stash P into per-wave LDS for A-fragment-layout re-read.
#pragma unroll
    for (int ct = 0; ct < 4; ++ct)
#pragma unroll
      for (int j = 0; j < 8; ++j)
        pl[wid][j + 8 * half][ct * 16 + l16] = sc[ct][j];
    __syncthreads();

    // O += P * V  (K = 64 keys of this chunk)
#pragma unroll
    for (int ft = 0; ft < 4; ++ft) {
      v8f o = of[ft];
#pragma unroll
      for (int t = 0; t < 16; ++t) {
        const int ka = 4 * t + 2 * half;
        v2f a, b;
        a.x = pl[wid][l16][ka];
        a.y = pl[wid][l16][ka + 1];
        b.x = vl[ka][ft * 16 + l16];      // B[kkey, nfeat] = v[kkey, nfeat]
        b.y = vl[ka + 1][ft * 16 + l16];
        o = wmma_f32x4(a, b, o);
      }
      of[ft] = o;
    }
  }

#pragma unroll
  for (int ft = 0; ft < 4; ++ft)
#pragma unroll
    for (int j = 0; j < 8; ++j) {
      const int r = bb * NSEQ + rowTile + j + 8 * half;
      const int col = head * DHEAD + ft * 16 + l16;
      O[(size_t)r * DIMD + col] = of[ft][j] / li[j];
    }
}

// ---------------------------------------------------------------------------
__global__ __launch_bounds__(256)
void rmsnorm_k(const float* __restrict__ X, const float* __restrict__ w,
               float* __restrict__ Y, int R) {
  const int wid  = threadIdx.x >> 5;
  const int lane = threadIdx.x & 31;
  const int row  = blockIdx.x * 8 + wid;
  if (row >= R) return;
  const float* x = X + (size_t)row * DIMD;
  float vals[16];
  float ss = 0.f;
#pragma unroll
  for (int i = 0; i < 16; ++i) {
    const float t = x[lane + 32 * i];
    vals[i] = t;
    ss += t * t;
  }
  ss = wave32_sum(ss);
  const float scale = rsqrtf(ss * (1.f / 512.f) + EPSF);
  float* y = Y + (size_t)row * DIMD;
#pragma unroll
  for (int i = 0; i < 16; ++i)
    y[lane + 32 * i] = vals[i] * scale * w[lane + 32 * i];
}

// ---------------------------------------------------------------------------
__global__ __launch_bounds__(256)
void ffact_k(const float* __restrict__ H, float* __restrict__ Aout) {
  const size_t i = (size_t)blockIdx.x * 256 + threadIdx.x;
  if (i >= (size_t)RTOT * DFF) return;
  const int r = (int)(i / DFF);
  const int c = (int)(i % DFF);
  const float s = H[(size_t)r * (2 * DFF) + c];
  const float g = H[(size_t)r * (2 * DFF) + DFF + c];
  const float gelu = 0.5f * g * (1.f + erff(g * 0.70710678118654752f));
  Aout[i] = s * gelu;
}

// ---------------------------------------------------------------------------
// Pooling attention: one block per query row (3072), wave w = head w.
// Query row r: qi=r/256, pos=r%256, message batch bc=qi/6 (reference quirk).
// ---------------------------------------------------------------------------
__global__ __launch_bounds__(256)
void pool_attn(const float* __restrict__ Qb, const float* __restrict__ KVp,
               float* __restrict__ Ob, int M) {
  const int r    = blockIdx.x;
  const int head = threadIdx.x >> 5;
  const int lane = threadIdx.x & 31;
  const int qi   = r >> 8;
  const int pos  = r & 255;
  const int bc   = qi / 6;

  const float* q = Qb + (size_t)r * DIMD + head * DHEAD;
  const float q0 = q[2 * lane], q1 = q[2 * lane + 1];

  float maxv = -3.402823466e+38f, sum = 0.f, o0 = 0.f, o1 = 0.f;
  for (int m = 0; m < M; ++m) {
    const float* kv = KVp + ((size_t)(m * 2 + bc) * NSEQ + pos) * 1024 + head * DHEAD;
    float p = q0 * kv[2 * lane] + q1 * kv[2 * lane + 1];
    p = wave32_sum(p);
    const float v0 = kv[512 + 2 * lane];
    const float v1 = kv[512 + 2 * lane + 1];
    if (p > maxv) {
      const float f = __expf(maxv - p);
      sum *= f; o0 *= f; o1 *= f;
      maxv = p;
    }
    const float e = __expf(p - maxv);
    sum += e; o0 += e * v0; o1 += e * v1;
  }
  float* o = Ob + (size_t)r * DIMD + head * DHEAD;
  o[2 * lane]     = o0 / sum;
  o[2 * lane + 1] = o1 / sum;
}

// ---------------------------------------------------------------------------
__global__ __launch_bounds__(256)
void bcast6_k(const float* __restrict__ src, float* __restrict__ dst) {
  const size_t i = (size_t)blockIdx.x * 256 + threadIdx.x;
  if (i < (size_t)NBLK * NBATCH * NSEQ * DIMD)
    dst[i] = src[i % ((size_t)NBATCH * NSEQ * DIMD)];
}

// ---------------------------------------------------------------------------
extern "C" void kernel_launch(void* const* d_in, const int* in_sizes, int n_in,
                              void* d_out, int out_size, void* d_ws, size_t ws_size,
                              hipStream_t stream) {
  (void)in_sizes; (void)n_in; (void)out_size; (void)ws_size;

  const float* tokens      = (const float*)d_in[0];
  const float* attn_norm_w = (const float*)d_in[1];
  const float* attn_wq     = (const float*)d_in[2];   // [512,512]
  const float* attn_wkv    = (const float*)d_in[3];   // [1024,512]
  const float* attn_wo     = (const float*)d_in[4];   // [512,512]
  const float* ff_norm_w   = (const float*)d_in[5];
  const float* ff_keys_w   = (const float*)d_in[6];   // [2730,512]
  const float* ff_keys_b   = (const float*)d_in[7];   // [2730]
  const float* ff_values_w = (const float*)d_in[8];   // [512,1365]
  const float* ff_values_b = (const float*)d_in[9];   // [512]
  const float* res_norm_w  = (const float*)d_in[10];
  const float* res_wq      = (const float*)d_in[11];  // [512,512]
  const float* res_wkv     = (const float*)d_in[12];  // [1024,512]
  const float* res_wo      = (const float*)d_in[13];  // [512,512]

  // workspace layout (floats)
  float* ws = (float*)d_ws;
  size_t off = 0;
  float* tok = ws + off; off += (size_t)RTOT * DIMD;          // current tokens
  float* xn  = ws + off; off += (size_t)RTOT * DIMD;          // rmsnorm scratch
  float* qb  = ws + off; off += (size_t)RTOT * DIMD;          // q projections
  float* kvb = ws + off; off += (size_t)RTOT * 1024;          // self-attn kv
  float* ao  = ws + off; off += (size_t)RTOT * DIMD;          // attn out (pre-Wo)
  float* att = ws + off; off += (size_t)RTOT * DIMD;          // attended
  float* ret = ws + off; off += (size_t)RTOT * DIMD;          // retrieved
  float* ffh = ws + off; off += (size_t)RTOT * (2 * DFF);     // ff hidden
  float* ffa = ws + off; off += (size_t)RTOT * DFF;           // ff activated
  float* kvp = ws + off; off += (size_t)42 * 512 * 1024;      // pooled-msg kv cache

  const dim3 blk256(256), blk128(128);
  const dim3 gN512(4, 48), gN1024(8, 48), gN2730(22, 48);

  bcast6_k<<<(RTOT * DIMD + 255) / 256, blk256, 0, stream>>>(tokens, tok);
  gemm_wmma<<<gN1024, blk256, 0, stream>>>(tok, res_wkv, nullptr, kvp, RTOT, 1024, DIMD);

  for (int e = 0; e < 3; ++e) {
    // ---- self attention over flat [12,256,512] ----
    rmsnorm_k<<<RTOT / 8, blk256, 0, stream>>>(tok, attn_norm_w, xn, RTOT);
    gemm_wmma<<<gN512, blk256, 0, stream>>>(xn, attn_wq, nullptr, qb, RTOT, 512, DIMD);
    gemm_wmma<<<gN1024, blk256, 0, stream>>>(tok, attn_wkv, nullptr, kvb, RTOT, 1024, DIMD); // kv from RAW x
    attn_self<<<12 * 8 * 4, blk128, 0, stream>>>(qb, kvb, ao);
    gemm_wmma<<<gN512, blk256, 0, stream>>>(ao, attn_wo, nullptr, att, RTOT, 512, DIMD);

    // ---- feedforward (GEGLU, exact gelu) ----
    rmsnorm_k<<<RTOT / 8, blk256, 0, stream>>>(tok, ff_norm_w, xn, RTOT);
    gemm_wmma<<<gN2730, blk256, 0, stream>>>(xn, ff_keys_w, ff_keys_b, ffh, RTOT, 2 * DFF, DIMD);
    ffact_k<<<(RTOT * DFF + 255) / 256, blk256, 0, stream>>>(ffh, ffa);
    gemm_wmma<<<gN512, blk256, 0, stream>>>(ffa, ff_values_w, ff_values_b, ret, RTOT, 512, DFF);

    // ---- append kv for the 12 new message entries ----
    gemm_wmma<<<gN1024, blk256, 0, stream>>>(att, res_wkv, nullptr,
        kvp + (size_t)(6 + 12 * e) * 512 * 1024, RTOT, 1024, DIMD);
    gemm_wmma<<<gN1024, blk256, 0, stream>>>(ret, res_wkv, nullptr,
        kvp + (size_t)(12 + 12 * e) * 512 * 1024, RTOT, 1024, DIMD);

    // ---- pooling attention over M = 18 + 12e messages ----
    rmsnorm_k<<<RTOT / 8, blk256, 0, stream>>>(tok, res_norm_w, xn, RTOT);
    gemm_wmma<<<gN512, blk256, 0, stream>>>(xn, res_wq, nullptr, qb, RTOT, 512, DIMD);
    pool_attn<<<RTOT, blk256, 0, stream>>>(qb, kvp, ao, 18 + 12 * e);
    float* dst = (e == 2) ? (float*)d_out : tok;
    gemm_wmma<<<gN512, blk256, 0, stream>>>(ao, res_wo, nullptr, dst, RTOT, 512, DIMD);
  }
}